// GCN4OPTIMAL_48679159333392
// MI455X (gfx1250) — compile-verified
//
#include <hip/hip_runtime.h>
#include <math.h>

typedef float v2f __attribute__((ext_vector_type(2)));
typedef float v8f __attribute__((ext_vector_type(8)));

#define HDIM 128
#define NCLS 10

// ---------------------------------------------------------------------------
// GEMM: out[R x HDIM] = in[R x K] * W[K x HDIM], all row-major fp32.
// Requires R % 16 == 0, K % 4 == 0 (true here: R=50000, K in {256,128}).
// One wave computes one 16x16 tile with V_WMMA_F32_16X16X4_F32.
// ---------------------------------------------------------------------------
__global__ void gcn_gemm_wmma_f32(const float* __restrict__ in,
                                  const float* __restrict__ W,
                                  float* __restrict__ out,
                                  int R, int K) {
  const int lane = threadIdx.x & 31;
  const int wave = threadIdx.x >> 5;
  const int tiles_n = HDIM / 16;                 // 8
  const int tiles_total = (R / 16) * tiles_n;
  const int tile = blockIdx.x * 8 + wave;        // 8 waves per 256-thread block
  if (tile >= tiles_total) return;               // uniform per-wave branch

  const int tm = tile / tiles_n;
  const int tn = tile - tm * tiles_n;
  const int row0 = tm * 16;
  const int col0 = tn * 16;
  const int l    = lane & 15;                    // M (for A) / N (for B,C,D)
  const int half = lane >> 4;                    // K-pair selector

  // A fragment base: lane l holds in[row0+l][k0 + 2*half + {0,1}] (float2)
  const float* __restrict__ arow = in + (size_t)(row0 + l) * K + 2 * half;
  // B fragment base: lane l holds W[k0 + 2*half + {0,1}][col0 + l]
  const float* __restrict__ bcol = W + (size_t)(2 * half) * HDIM + col0 + l;

  v8f c = {0.f, 0.f, 0.f, 0.f, 0.f, 0.f, 0.f, 0.f};
  #pragma unroll 4
  for (int k0 = 0; k0 < K; k0 += 4) {
    v2f a = *(const v2f*)(arow + k0);
    v2f b;
    b.x = bcol[(size_t)k0 * HDIM];
    b.y = bcol[(size_t)(k0 + 1) * HDIM];
    // D = A(16x4) * B(4x16) + C   (8 args: neg_a, A, neg_b, B, c_mod, C, reuse_a, reuse_b)
    c = __builtin_amdgcn_wmma_f32_16x16x4_f32(false, a, false, b, (short)0, c,
                                              false, false);
  }

  // C/D layout: VGPR r -> row (r + 8*half), col = col0 + l
  float* __restrict__ orow = out + (size_t)(row0 + 8 * half) * HDIM + col0 + l;
  #pragma unroll
  for (int r = 0; r < 8; ++r) orow[(size_t)r * HDIM] = c[r];
}

// ---------------------------------------------------------------------------
// Degree / normalization
// ---------------------------------------------------------------------------
__global__ void gcn_deg_init(float* __restrict__ deg, int N) {
  int i = blockIdx.x * blockDim.x + threadIdx.x;
  if (i < N) deg[i] = 1.0f;                      // self-loop weight
}

__global__ void gcn_deg_accum(const int* __restrict__ dst,
                              const float* __restrict__ ew,
                              float* __restrict__ deg, int E) {
  int e = blockIdx.x * blockDim.x + threadIdx.x;
  if (e < E) unsafeAtomicAdd(&deg[dst[e]], ew[e]);
}

__global__ void gcn_deg_finalize(float* __restrict__ deg, int N) {
  int i = blockIdx.x * blockDim.x + threadIdx.x;
  if (i < N) {
    float d = deg[i];
    deg[i] = d > 0.f ? rsqrtf(d) : 0.f;          // becomes dinv in place
  }
}

// ---------------------------------------------------------------------------
// agg[i][:] = dinv[i]^2 * h[i][:]   (self-loop contribution, initializes agg)
// one wave per node, 4 floats per lane
// ---------------------------------------------------------------------------
__global__ void gcn_agg_init(const float* __restrict__ dinv,
                             const float* __restrict__ h,
                             float* __restrict__ agg, int N) {
  int t = blockIdx.x * blockDim.x + threadIdx.x;
  int i = t >> 5;
  if (i >= N) return;
  int lane = t & 31;
  float c = dinv[i] * dinv[i];
  const float4 hv = *(const float4*)(h + (size_t)i * HDIM + lane * 4);
  float4 o;
  o.x = c * hv.x; o.y = c * hv.y; o.z = c * hv.z; o.w = c * hv.w;
  *(float4*)(agg + (size_t)i * HDIM + lane * 4) = o;
}

// ---------------------------------------------------------------------------
// Edge scatter: agg[dst] += (dinv[src]*w*dinv[dst]) * h[src]
// one wave per edge, 4 floats per lane, hw fp32 atomics (L2-resident)
// ---------------------------------------------------------------------------
__global__ void gcn_scatter(const int* __restrict__ src,
                            const int* __restrict__ dst,
                            const float* __restrict__ ew,
                            const float* __restrict__ dinv,
                            const float* __restrict__ h,
                            float* __restrict__ agg, int E) {
  int t = blockIdx.x * blockDim.x + threadIdx.x;
  int e = t >> 5;
  if (e >= E) return;
  int lane = t & 31;
  int s = src[e], d = dst[e];
  float coef = dinv[s] * ew[e] * dinv[d];
  const float4 hv = *(const float4*)(h + (size_t)s * HDIM + lane * 4);
  float* ap = agg + (size_t)d * HDIM + lane * 4;
  unsafeAtomicAdd(ap + 0, coef * hv.x);
  unsafeAtomicAdd(ap + 1, coef * hv.y);
  unsafeAtomicAdd(ap + 2, coef * hv.z);
  unsafeAtomicAdd(ap + 3, coef * hv.w);
}

// ---------------------------------------------------------------------------
// In-place bias + ReLU on [N x HDIM]
// ---------------------------------------------------------------------------
__global__ void gcn_bias_relu(float* __restrict__ a,
                              const float* __restrict__ b, int total) {
  int t = blockIdx.x * blockDim.x + threadIdx.x;
  if (t >= total) return;
  int f = t & (HDIM - 1);
  a[t] = fmaxf(a[t] + b[f], 0.f);
}

// ---------------------------------------------------------------------------
// Head: logits = h @ fc_w + fc_b ; softmax (max-subtracted). One thread/row.
// ---------------------------------------------------------------------------
__global__ void gcn_head_softmax(const float* __restrict__ h,
                                 const float* __restrict__ fcw,
                                 const float* __restrict__ fcb,
                                 float* __restrict__ out, int N) {
  int i = blockIdx.x * blockDim.x + threadIdx.x;
  if (i >= N) return;
  float logit[NCLS];
  #pragma unroll
  for (int c = 0; c < NCLS; ++c) logit[c] = fcb[c];
  const float* hr = h + (size_t)i * HDIM;
  for (int f = 0; f < HDIM; ++f) {
    float hv = hr[f];
    #pragma unroll
    for (int c = 0; c < NCLS; ++c) logit[c] = fmaf(hv, fcw[f * NCLS + c], logit[c]);
  }
  float m = logit[0];
  #pragma unroll
  for (int c = 1; c < NCLS; ++c) m = fmaxf(m, logit[c]);
  float ex[NCLS], sum = 0.f;
  #pragma unroll
  for (int c = 0; c < NCLS; ++c) { ex[c] = __expf(logit[c] - m); sum += ex[c]; }
  float inv = 1.f / sum;
  float* orow = out + (size_t)i * NCLS;
  #pragma unroll
  for (int c = 0; c < NCLS; ++c) orow[c] = ex[c] * inv;
}

// ---------------------------------------------------------------------------
extern "C" void kernel_launch(void* const* d_in, const int* in_sizes, int n_in,
                              void* d_out, int out_size, void* d_ws, size_t ws_size,
                              hipStream_t stream) {
  const float* x   = (const float*)d_in[0];
  const int*   ei  = (const int*)  d_in[1];
  const float* ew  = (const float*)d_in[2];
  const float* W1  = (const float*)d_in[3];
  const float* b1  = (const float*)d_in[4];
  const float* W2  = (const float*)d_in[5];
  const float* b2  = (const float*)d_in[6];
  const float* W3  = (const float*)d_in[7];
  const float* b3  = (const float*)d_in[8];
  const float* fcw = (const float*)d_in[9];
  const float* fcb = (const float*)d_in[10];

  const int F = 256;
  const int N = in_sizes[0] / F;
  const int E = in_sizes[1] / 2;
  const int* src = ei;
  const int* dst = ei + E;

  // Workspace carve-up: dinv[N] | A[N*128] | B[N*128]
  char* ws = (char*)d_ws;
  float* dinv = (float*)ws;
  size_t off = (((size_t)N * 4) + 255) & ~(size_t)255;
  float* A = (float*)(ws + off);
  size_t msz = (((size_t)N * HDIM * 4) + 255) & ~(size_t)255;
  float* B = (float*)(ws + off + msz);

  const int TB = 256;

  // --- normalization coefficients ---
  gcn_deg_init    <<<(N + TB - 1) / TB, TB, 0, stream>>>(dinv, N);
  gcn_deg_accum   <<<(E + TB - 1) / TB, TB, 0, stream>>>(dst, ew, dinv, E);
  gcn_deg_finalize<<<(N + TB - 1) / TB, TB, 0, stream>>>(dinv, N);

  const int tiles  = (N / 16) * (HDIM / 16);
  const int gblk   = (tiles + 7) / 8;               // 8 waves (tiles) per block
  const int nwarpN = ((N * 32) + TB - 1) / TB;      // wave-per-node grids
  const int nwarpE = (int)((((size_t)E * 32) + TB - 1) / TB);
  const int elemN  = ((N * HDIM) + TB - 1) / TB;

  // --- layer 1: in = x (K=256) ---
  gcn_gemm_wmma_f32<<<gblk, TB, 0, stream>>>(x, W1, A, N, F);
  gcn_agg_init     <<<nwarpN, TB, 0, stream>>>(dinv, A, B, N);
  gcn_scatter      <<<nwarpE, TB, 0, stream>>>(src, dst, ew, dinv, A, B, E);
  gcn_bias_relu    <<<elemN, TB, 0, stream>>>(B, b1, N * HDIM);   // h1 in B

  // --- layer 2: in = B (K=128) ---
  gcn_gemm_wmma_f32<<<gblk, TB, 0, stream>>>(B, W2, A, N, HDIM);
  gcn_agg_init     <<<nwarpN, TB, 0, stream>>>(dinv, A, B, N);
  gcn_scatter      <<<nwarpE, TB, 0, stream>>>(src, dst, ew, dinv, A, B, E);
  gcn_bias_relu    <<<elemN, TB, 0, stream>>>(B, b2, N * HDIM);   // h2 in B

  // --- layer 3: in = B (K=128) ---
  gcn_gemm_wmma_f32<<<gblk, TB, 0, stream>>>(B, W3, A, N, HDIM);
  gcn_agg_init     <<<nwarpN, TB, 0, stream>>>(dinv, A, B, N);
  gcn_scatter      <<<nwarpE, TB, 0, stream>>>(src, dst, ew, dinv, A, B, E);
  gcn_bias_relu    <<<elemN, TB, 0, stream>>>(B, b3, N * HDIM);   // h3 in B

  // --- head + softmax ---
  gcn_head_softmax<<<(N + TB - 1) / TB, TB, 0, stream>>>(B, fcw, fcb,
                                                         (float*)d_out, N);
}